// GCN_2585570312415
// MI455X (gfx1250) — compile-verified
//
#include <hip/hip_runtime.h>

typedef __attribute__((ext_vector_type(2))) float v2f;
typedef __attribute__((ext_vector_type(8))) float v8f;

#define IN_F 128
#define HID  64
#define PSTRIDE 80   // pair-row stride in v2f units: 640B -> +32 bank rotation per row,
                     // so lanes 0-15 (row p) and 16-31 (row p+1) use disjoint bank halves

// ---------------- init: deg=1 (self loop), agg1=0, out=0 ----------------
__global__ void gcn_init(float* __restrict__ deg, float* __restrict__ agg1,
                         float* __restrict__ out, int n) {
  long i = (long)blockIdx.x * blockDim.x + threadIdx.x;
  long tot = (long)n * HID;
  if (i < tot) agg1[i] = 0.0f;
  if (i < n) deg[i] = 1.0f;               // self-loop contributes 1 to degree
  if (i < 2L * n) out[i] = 0.0f;
}

// ---------------- degree accumulation on dst ----------------
__global__ void gcn_degree(const int* __restrict__ dst, float* __restrict__ deg, int E) {
  long e = (long)blockIdx.x * blockDim.x + threadIdx.x;
  if (e < E) unsafeAtomicAdd(&deg[dst[e]], 1.0f);
}

// ---------------- dinv = rsqrt(deg), in place ----------------
__global__ void gcn_rsqrt(float* __restrict__ dinv, int n) {
  long i = (long)blockIdx.x * blockDim.x + threadIdx.x;
  if (i < n) dinv[i] = rsqrtf(dinv[i]);
}

// ---------------- layer-1 GEMM via fp32 WMMA: hs = (x @ W1) * dinv[row] ----------------
// Each wave: one 16-row M tile, full N=64 (4 accumulator tiles), K=128 in steps of 4.
// W1 staged in LDS as K-pairs: sP[p*PSTRIDE + n] = {W1[2p][n], W1[2p+1][n]}, so every
// B fragment is one aligned ds_load_b64 into an even VGPR pair (no repack moves).
__global__ void __launch_bounds__(128)
gcn_gemm1_wmma(const float* __restrict__ x, const float* __restrict__ W1,
               const float* __restrict__ dinv, float* __restrict__ hs,
               int n, int mtiles) {
  __shared__ v2f sP[(IN_F / 2) * PSTRIDE];               // 40 KB of the 320 KB/WGP
  for (int i = threadIdx.x; i < IN_F * HID; i += 128) {
    int k  = i >> 6;                                     // i / HID
    int nn = i & (HID - 1);
    ((float*)sP)[(((k >> 1) * PSTRIDE + nn) << 1) + (k & 1)] = W1[i];
  }
  __syncthreads();

  const int wave = threadIdx.x >> 5;
  const int lane = threadIdx.x & 31;
  const int tile = blockIdx.x * 4 + wave;
  if (tile >= mtiles) return;

  const int m0    = tile * 16;
  const int col   = lane & 15;             // N (for B/C/D) or M (for A) position
  const int half  = lane >> 4;             // 0: K=0,1  1: K=2,3 of each 4-step
  const int khalf = half * 2;

  v8f acc0 = {}, acc1 = {}, acc2 = {}, acc3 = {};

  int arow = m0 + col; if (arow >= n) arow = n - 1;      // clamp (n is a multiple of 16 anyway)
  const float* xrow = x + (long)arow * IN_F;

  for (int k0 = 0; k0 < IN_F; k0 += 4) {
    // A fragment: 16x4 fp32 — lane holds K = khalf, khalf+1 of row (lane&15)
    v2f a = *reinterpret_cast<const v2f*>(xrow + k0 + khalf);
    const v2f* bp = sP + ((k0 >> 1) + half) * PSTRIDE + col;
    v2f b0 = bp[0];
    v2f b1 = bp[16];
    v2f b2 = bp[32];
    v2f b3 = bp[48];
    acc0 = __builtin_amdgcn_wmma_f32_16x16x4_f32(false, a, false, b0, (short)0, acc0, false, false);
    acc1 = __builtin_amdgcn_wmma_f32_16x16x4_f32(false, a, false, b1, (short)0, acc1, false, false);
    acc2 = __builtin_amdgcn_wmma_f32_16x16x4_f32(false, a, false, b2, (short)0, acc2, false, false);
    acc3 = __builtin_amdgcn_wmma_f32_16x16x4_f32(false, a, false, b3, (short)0, acc3, false, false);
  }

  // C/D layout: VGPR r holds M = r (lanes 0-15) or M = r+8 (lanes 16-31), N = lane&15.
  const int rbase = m0 + (half << 3);
  for (int r = 0; r < 8; ++r) {
    int row = rbase + r;
    if (row < n) {
      float w = dinv[row];                               // pre-fold dinv[src]
      float* o = hs + (long)row * HID + col;
      o[ 0] = acc0[r] * w;
      o[16] = acc1[r] * w;
      o[32] = acc2[r] * w;
      o[48] = acc3[r] * w;
    }
  }
}

// ---------------- layer-1 edge scatter: agg1[dst] += hs[src] * dinv[dst] ----------------
__global__ void gcn_edge1(const int* __restrict__ src, const int* __restrict__ dst,
                          const float* __restrict__ dinv, const float* __restrict__ hs,
                          float* __restrict__ agg1, int E) {
  long t = (long)blockIdx.x * blockDim.x + threadIdx.x;
  if (t >= (long)E * 16) return;
  int e = (int)(t >> 4);
  int c = ((int)t & 15) * 4;
  int s = src[e], d = dst[e];
  float w = dinv[d];
  const float4 hv = *reinterpret_cast<const float4*>(hs + (long)s * HID + c);
  float* o = agg1 + (long)d * HID + c;
  unsafeAtomicAdd(o + 0, hv.x * w);
  unsafeAtomicAdd(o + 1, hv.y * w);
  unsafeAtomicAdd(o + 2, hv.z * w);
  unsafeAtomicAdd(o + 3, hv.w * w);
}

// ---------------- layer-1 finalize: + self-loop + b1, ReLU (in place) ----------------
__global__ void gcn_fin1(float* __restrict__ agg1, const float* __restrict__ hs,
                         const float* __restrict__ dinv, const float* __restrict__ b1, int n) {
  long i = (long)blockIdx.x * blockDim.x + threadIdx.x;
  if (i >= (long)n * HID) return;
  int node = (int)(i >> 6);
  int f    = (int)i & (HID - 1);
  float v = agg1[i] + hs[i] * dinv[node] + b1[f];        // hs already carries dinv[src]
  agg1[i] = v > 0.0f ? v : 0.0f;
}

// ---------------- layer-2 GEMM (64 -> 2), pre-scaled by dinv ----------------
__global__ void gcn_gemm2(const float* __restrict__ h, const float* __restrict__ W2,
                          const float* __restrict__ dinv, float* __restrict__ gs, int n) {
  long i = (long)blockIdx.x * blockDim.x + threadIdx.x;
  if (i >= n) return;
  const float* hrow = h + i * HID;
  float a0 = 0.0f, a1 = 0.0f;
#pragma unroll
  for (int f = 0; f < HID; ++f) {
    float hv = hrow[f];
    a0 += hv * W2[2 * f + 0];
    a1 += hv * W2[2 * f + 1];
  }
  float w = dinv[i];
  gs[2 * i + 0] = a0 * w;
  gs[2 * i + 1] = a1 * w;
}

// ---------------- layer-2 edge scatter ----------------
__global__ void gcn_edge2(const int* __restrict__ src, const int* __restrict__ dst,
                          const float* __restrict__ dinv, const float* __restrict__ gs,
                          float* __restrict__ out, int E) {
  long e = (long)blockIdx.x * blockDim.x + threadIdx.x;
  if (e >= E) return;
  int s = src[e], d = dst[e];
  float w = dinv[d];
  unsafeAtomicAdd(&out[2L * d + 0], gs[2L * s + 0] * w);
  unsafeAtomicAdd(&out[2L * d + 1], gs[2L * s + 1] * w);
}

// ---------------- layer-2 finalize: + self-loop + b2 ----------------
__global__ void gcn_fin2(float* __restrict__ out, const float* __restrict__ gs,
                         const float* __restrict__ dinv, const float* __restrict__ b2, int n) {
  long i = (long)blockIdx.x * blockDim.x + threadIdx.x;
  if (i >= 2L * n) return;
  int node = (int)(i >> 1);
  int c    = (int)i & 1;
  out[i] += gs[i] * dinv[node] + b2[c];
}

extern "C" void kernel_launch(void* const* d_in, const int* in_sizes, int n_in,
                              void* d_out, int out_size, void* d_ws, size_t ws_size,
                              hipStream_t stream) {
  const float* x  = (const float*)d_in[0];
  const int*   ei = (const int*)d_in[1];           // [2, E] flattened: row0=src, row1=dst
  const float* W1 = (const float*)d_in[2];
  const float* b1 = (const float*)d_in[3];
  const float* W2 = (const float*)d_in[4];
  const float* b2 = (const float*)d_in[5];
  float* out = (float*)d_out;

  const int n = in_sizes[0] / IN_F;
  const int E = in_sizes[1] / 2;
  const int* src = ei;
  const int* dst = ei + E;

  // workspace layout (floats): dinv[n] | hs[64n] | agg1[64n] | gs[2n]
  float* ws   = (float*)d_ws;
  float* dinv = ws;
  float* hs   = ws + (long)n;
  float* agg1 = ws + (long)n * 65;
  float* gs   = ws + (long)n * 129;

  const long nh = (long)n * HID;

  gcn_init  <<<(unsigned)((nh + 255) / 256),        256, 0, stream>>>(dinv, agg1, out, n);
  gcn_degree<<<(unsigned)(((long)E + 255) / 256),   256, 0, stream>>>(dst, dinv, E);
  gcn_rsqrt <<<(unsigned)(((long)n + 255) / 256),   256, 0, stream>>>(dinv, n);

  const int mtiles = (n + 15) / 16;
  gcn_gemm1_wmma<<<(unsigned)((mtiles + 3) / 4),    128, 0, stream>>>(x, W1, dinv, hs, n, mtiles);

  const long t1 = (long)E * 16;
  gcn_edge1 <<<(unsigned)((t1 + 255) / 256),        256, 0, stream>>>(src, dst, dinv, hs, agg1, E);
  gcn_fin1  <<<(unsigned)((nh + 255) / 256),        256, 0, stream>>>(agg1, hs, dinv, b1, n);
  gcn_gemm2 <<<(unsigned)(((long)n + 255) / 256),   256, 0, stream>>>(agg1, W2, dinv, gs, n);
  gcn_edge2 <<<(unsigned)(((long)E + 255) / 256),   256, 0, stream>>>(src, dst, dinv, gs, out, E);
  gcn_fin2  <<<(unsigned)((2L * n + 255) / 256),    256, 0, stream>>>(out, gs, dinv, b2, n);
}